// RegionExtractor_68332929679632
// MI455X (gfx1250) — compile-verified
//
#include <hip/hip_runtime.h>
#include <stdint.h>

#define B_      4
#define N_      1000
#define PAD_    1024
#define MAXD_   100
#define HW_     65536                      // 256*256 floats per mask
#define CHUNKS_ 4
#define CHUNK_ELEMS_ (HW_ / CHUNKS_)       // 16384 floats per block (64 KB)
#define ROUNDS_ (CHUNK_ELEMS_ / 4 / 256)   // 16 float4 rounds per block
#define MIN_SIZE_ 25.0f
#define IOU_T_  0.3f

// d_out layout (float elements), outputs concatenated in reference return order:
// boxes [4,100,4], masks [4,100,1,256,256], scores [4,100], labels [4,100], valid [4,100]
static constexpr size_t OUT_BOXES_  = 0;
static constexpr size_t OUT_MASKS_  = (size_t)B_ * MAXD_ * 4;                 // 1600
static constexpr size_t OUT_SCORES_ = OUT_MASKS_ + (size_t)B_ * MAXD_ * HW_;  // 26216000
static constexpr size_t OUT_LABELS_ = OUT_SCORES_ + (size_t)B_ * MAXD_;
static constexpr size_t OUT_VALID_  = OUT_LABELS_ + (size_t)B_ * MAXD_;

// ---------------------------------------------------------------------------
// Kernel 1: one workgroup (1024 threads = 32 wave32) per batch image.
// Sort -> IoU bitmask matrix in LDS -> single-wave greedy NMS -> top-100 ids.
// ---------------------------------------------------------------------------
__global__ __launch_bounds__(PAD_) void nms_kernel(const float* __restrict__ boxes,
                                                   const float* __restrict__ scores,
                                                   int* __restrict__ ws_fi,
                                                   int* __restrict__ ws_v) {
  __shared__ float    skey[PAD_];
  __shared__ int      sidx[PAD_];
  __shared__ float    sx0[PAD_], sy0[PAD_], sx1[PAD_], sy1[PAD_], sar[PAD_];
  __shared__ int      svalid[PAD_];
  __shared__ unsigned smat[N_ * 32];       // row i: suppression bitmask over 1024 sorted cols

  const int tid = threadIdx.x;
  const int bb  = blockIdx.x;

  // init selection output (deterministic every call)
  if (tid < MAXD_) { ws_fi[bb * MAXD_ + tid] = -1; ws_v[bb * MAXD_ + tid] = 0; }

  // key = score if area > MIN_SIZE else -inf; pad slots -inf
  float key = -3.0e38f;
  if (tid < N_) {
    const float* bp = boxes + ((size_t)bb * N_ + tid) * 4;
    float area = (bp[2] - bp[0]) * (bp[3] - bp[1]);
    if (area > MIN_SIZE_) key = scores[(size_t)bb * N_ + tid];
  }
  skey[tid] = key;
  sidx[tid] = tid;
  __syncthreads();

  // bitonic sort (descending by key), index payload
  for (int k = 2; k <= PAD_; k <<= 1) {
    for (int j = k >> 1; j > 0; j >>= 1) {
      int ixj = tid ^ j;
      if (ixj > tid) {
        float ka = skey[tid], kb = skey[ixj];
        bool up = ((tid & k) == 0);
        if (up ? (ka < kb) : (ka > kb)) {
          skey[tid] = kb; skey[ixj] = ka;
          int t0 = sidx[tid]; sidx[tid] = sidx[ixj]; sidx[ixj] = t0;
        }
      }
      __syncthreads();
    }
  }

  // gather sorted geometry (invalid slots zeroed; they can never be 'keep')
  {
    float x0 = 0.f, y0 = 0.f, x1 = 0.f, y1 = 0.f, ar = 0.f; int vv = 0;
    if (skey[tid] > -1.0e30f) {
      int oi = sidx[tid];
      const float* bp = boxes + ((size_t)bb * N_ + oi) * 4;
      x0 = bp[0]; y0 = bp[1]; x1 = bp[2]; y1 = bp[3];
      ar = (x1 - x0) * (y1 - y0); vv = 1;
    }
    sx0[tid] = x0; sy0[tid] = y0; sx1[tid] = x1; sy1[tid] = y1;
    sar[tid] = ar; svalid[tid] = vv;
  }
  __syncthreads();

  // build (iou > thresh && j > i) bitmask matrix: 32000 u32 chunks over 1024 threads
  for (int t = tid; t < N_ * 32; t += PAD_) {
    int i = t >> 5, c = t & 31;
    unsigned m = 0u;
    if ((c << 5) + 31 > i) {                 // chunk holds some column j > i
      float ix0 = sx0[i], iy0 = sy0[i], ix1 = sx1[i], iy1 = sy1[i], ia = sar[i];
      #pragma unroll
      for (int u = 0; u < 32; ++u) {
        int j = (c << 5) + u;
        if (j > i) {
          float w = fminf(ix1, sx1[j]) - fmaxf(ix0, sx0[j]);
          float h = fminf(iy1, sy1[j]) - fmaxf(iy0, sy0[j]);
          w = fmaxf(w, 0.f); h = fmaxf(h, 0.f);
          float inter = w * h;
          float den = ia + sar[j] - inter;
          // iou > T  <=>  inter > T*den (den > 0); avoids the divide entirely
          if (den > 0.f && inter > IOU_T_ * den) m |= (1u << u);
        }
      }
    }
    smat[(i << 5) + c] = m;
  }
  __syncthreads();

  // single-wave greedy NMS: lane c owns suppression bits for columns [32c, 32c+31]
  if (tid < 32) {
    unsigned sup = 0u;
    for (int i = 0; i < N_; ++i) {
      unsigned wi = __shfl(sup, i >> 5, 32);
      int keep = svalid[i] && !((wi >> (i & 31)) & 1u);
      if (keep) sup |= smat[(i << 5) + tid];   // wave-uniform branch
    }
    // keep mask for this lane's 32 sorted positions
    unsigned km = 0u;
    for (int u = 0; u < 32; ++u) {
      int j = (tid << 5) + u;
      if (svalid[j] && !((sup >> u) & 1u)) km |= (1u << u);
    }
    // exclusive prefix of keep-counts across lanes
    int pc = __popc(km);
    int pre = 0;
    for (int l = 0; l < 32; ++l) {
      int o = __shfl(pc, l, 32);
      if (l < tid) pre += o;
    }
    // emit first MAX_DET survivors (already score-ordered)
    int r = pre;
    for (int u = 0; u < 32 && r < MAXD_; ++u) {
      if ((km >> u) & 1u) {
        ws_fi[bb * MAXD_ + r] = sidx[(tid << 5) + u];  // original index
        ws_v[bb * MAXD_ + r]  = 1;
        ++r;
      }
    }
  }
}

// ---------------------------------------------------------------------------
// Kernel 2: scalar outputs (boxes / scores / labels / valid)
// ---------------------------------------------------------------------------
__global__ void emit_small_kernel(const float* __restrict__ boxes,
                                  const float* __restrict__ scores,
                                  const int* __restrict__ labels,
                                  const int* __restrict__ ws_fi,
                                  const int* __restrict__ ws_v,
                                  float* __restrict__ out) {
  int bb = blockIdx.x, d = threadIdx.x;
  if (d >= MAXD_) return;
  int slot = bb * MAXD_ + d;
  int fi = ws_fi[slot], v = ws_v[slot];
  float b0 = 0.f, b1 = 0.f, b2 = 0.f, b3 = 0.f, sc = 0.f, lb = -1.f, vo = 0.f;
  if (v && fi >= 0) {
    const float* bp = boxes + ((size_t)bb * N_ + fi) * 4;
    b0 = bp[0]; b1 = bp[1]; b2 = bp[2]; b3 = bp[3];
    sc = scores[(size_t)bb * N_ + fi];
    lb = (float)labels[(size_t)bb * N_ + fi];
    vo = 1.f;
  }
  float* ob = out + OUT_BOXES_ + (size_t)slot * 4;
  ob[0] = b0; ob[1] = b1; ob[2] = b2; ob[3] = b3;
  out[OUT_SCORES_ + slot] = sc;
  out[OUT_LABELS_ + slot] = lb;
  out[OUT_VALID_  + slot] = vo;
}

// ---------------------------------------------------------------------------
// Kernel 3: bandwidth-bound mask gather through the CDNA5 async LDS engine,
// now fully pipelined: stage the whole 64KB chunk in LDS with 16 back-to-back
// async loads (ASYNCcnt=16 in flight -> latency amortized), one wait, then 16
// back-to-back async stores, one wait. Two exposed latencies per block instead
// of 32 -> copy runs at the ~23.3 TB/s HBM roofline, not the latency bound.
// 64KB LDS/block still allows 5 concurrent workgroups per 320KB WGP.
// ---------------------------------------------------------------------------
__global__ __launch_bounds__(256) void mask_copy_kernel(const float* __restrict__ masks,
                                                        const int* __restrict__ ws_fi,
                                                        const int* __restrict__ ws_v,
                                                        float* __restrict__ out) {
  __shared__ float4 stage[ROUNDS_ * 256];     // 16 * 256 * 16B = 64 KB
  const int blk   = blockIdx.x;
  const int chunk = blk % CHUNKS_;
  const int slot  = blk / CHUNKS_;            // b*MAXD + d
  const int bb    = slot / MAXD_;
  const int tid   = threadIdx.x;

  const int fi = ws_fi[slot], v = ws_v[slot];
  float* dst = out + OUT_MASKS_ + (size_t)slot * HW_ + (size_t)chunk * CHUNK_ELEMS_;

  if (!(v && fi >= 0)) {                      // block-uniform: zero fill invalid slots
    float4 z = make_float4(0.f, 0.f, 0.f, 0.f);
    for (int t = tid; t < CHUNK_ELEMS_ / 4; t += 256) ((float4*)dst)[t] = z;
    return;
  }

  const float* src = masks + ((size_t)bb * N_ + (size_t)fi) * HW_
                           + (size_t)chunk * CHUNK_ELEMS_;
  // low 32 bits of a generic pointer to LDS == workgroup-relative LDS offset
  const unsigned lbase = (unsigned)(uintptr_t)(&stage[0]);

  // phase 1: issue all async global->LDS loads (16 outstanding per wave)
  #pragma unroll
  for (int r = 0; r < ROUNDS_; ++r) {
    int idx = (r << 8) + tid;                 // float4 index within chunk
    unsigned lds = lbase + (unsigned)idx * 16u;
    unsigned long long gsrc = (unsigned long long)(uintptr_t)(src + (size_t)idx * 4);
    asm volatile("global_load_async_to_lds_b128 %0, %1, off"
                 :: "v"(lds), "v"(gsrc) : "memory");
  }
  asm volatile("s_wait_asynccnt 0" ::: "memory");   // all LDS slots written

  // phase 2: issue all async LDS->global stores
  #pragma unroll
  for (int r = 0; r < ROUNDS_; ++r) {
    int idx = (r << 8) + tid;
    unsigned lds = lbase + (unsigned)idx * 16u;
    unsigned long long gdst = (unsigned long long)(uintptr_t)(dst + (size_t)idx * 4);
    asm volatile("global_store_async_from_lds_b128 %0, %1, off"
                 :: "v"(gdst), "v"(lds) : "memory");
  }
  asm volatile("s_wait_asynccnt 0" ::: "memory");   // stores complete
}

// ---------------------------------------------------------------------------
extern "C" void kernel_launch(void* const* d_in, const int* in_sizes, int n_in,
                              void* d_out, int out_size, void* d_ws, size_t ws_size,
                              hipStream_t stream) {
  const float* boxes  = (const float*)d_in[0];   // [4,1000,4]
  const float* masks  = (const float*)d_in[1];   // [4,1000,1,256,256]
  const float* scores = (const float*)d_in[2];   // [4,1000]
  const int*   labels = (const int*)d_in[3];     // [4,1000] (int)
  float* out = (float*)d_out;

  int* ws_fi = (int*)d_ws;                       // [4*100] selected original index
  int* ws_v  = ws_fi + B_ * MAXD_;               // [4*100] valid flag

  nms_kernel<<<B_, PAD_, 0, stream>>>(boxes, scores, ws_fi, ws_v);
  emit_small_kernel<<<B_, MAXD_, 0, stream>>>(boxes, scores, labels, ws_fi, ws_v, out);
  mask_copy_kernel<<<B_ * MAXD_ * CHUNKS_, 256, 0, stream>>>(masks, ws_fi, ws_v, out);
}